// Voxelization_57062935495023
// MI455X (gfx1250) — compile-verified
//
#include <hip/hip_runtime.h>
#include <hip/hip_bf16.h>
#include <stdint.h>

// Problem constants (from reference): B=8, N=65536, C=32, R=64
#define VB 8
#define VN 65536
#define VC 32
#define VR 64
#define VR3 (VR * VR * VR)   // 262144
#define VEPS 1e-8f

// ---------------------------------------------------------------------------
// Kernel 1: per-(batch, axis) min over N points.
// 768 blocks (32 segments per (b,axis)); bit-cast atomicMin (values >= 0).
// minbits must be pre-set to 0x7F7F7F7F (= 3.39e38f) by the host memset.
// ---------------------------------------------------------------------------
__global__ __launch_bounds__(256) void k_min(const float* __restrict__ pts,
                                             unsigned* __restrict__ minbits) {
    __shared__ float red[256];
    const int t   = threadIdx.x;
    const int ba  = blockIdx.x >> 5;          // 0..23 : b*3 + axis
    const int seg = blockIdx.x & 31;          // 0..31 : 2048-elem segment
    const float* p = pts + (size_t)ba * VN + seg * 2048;
    float m = 3.4e38f;
#pragma unroll
    for (int i = 0; i < 8; ++i) m = fminf(m, p[t + i * 256]);
    red[t] = m;
    __syncthreads();
    for (int s = 128; s > 0; s >>= 1) {
        if (t < s) red[t] = fminf(red[t], red[t + s]);
        __syncthreads();
    }
    if (t == 0) atomicMin(&minbits[ba], __float_as_uint(red[0]));
}

// ---------------------------------------------------------------------------
// Kernel 2: global max of squared L2 norms of shifted points.
// 2048 blocks x 256 threads; one bit-cast atomicMax per block (values >= 0).
// ---------------------------------------------------------------------------
__global__ __launch_bounds__(256) void k_normmax(const float* __restrict__ pts,
                                                 const float* __restrict__ mins,
                                                 unsigned* __restrict__ nmaxsq_bits) {
    __shared__ float red[256];
    const int t = threadIdx.x;
    const int g = blockIdx.x * 256 + t;          // 0 .. B*N-1
    const int b = g >> 16;
    const int n = g & (VN - 1);
    const float* p = pts + (size_t)b * 3 * VN;
    const float x = p[n]          - mins[b * 3 + 0];
    const float y = p[VN + n]     - mins[b * 3 + 1];
    const float z = p[2 * VN + n] - mins[b * 3 + 2];
    red[t] = x * x + y * y + z * z;
    __syncthreads();
    for (int s = 128; s > 0; s >>= 1) {
        if (t < s) red[t] = fmaxf(red[t], red[t + s]);
        __syncthreads();
    }
    if (t == 0) atomicMax(nmaxsq_bits, __float_as_uint(red[0]));
}

// ---------------------------------------------------------------------------
// Kernel 3: scatter-add of features into d_out ([B,C,R^3] layout) + counts.
// Launched PER BATCH (256 blocks) so the 32 MB sum slab + 1 MB counts stay
// resident in the 192 MB L2 while its 2M scattered atomics execute.
// Each block handles 256 consecutive points; its 32x256-float feat tile
// (32 KB) is staged into LDS with CDNA5 async global->LDS loads (ASYNCcnt
// path), overlapped with the voxel-index ALU work.
// ---------------------------------------------------------------------------
__global__ __launch_bounds__(256) void k_accum(const float* __restrict__ pts,
                                               const float* __restrict__ feat,
                                               const float* __restrict__ mins,
                                               const unsigned* __restrict__ nmaxsq_bits,
                                               float* __restrict__ out,
                                               unsigned* __restrict__ cnt,
                                               int b) {
    __shared__ float tile[VC * 256];             // 32 KB: tile[c*256 + point]
    const int t  = threadIdx.x;
    const int n0 = blockIdx.x << 8;              // first point of this block

    // ---- issue async copies: feat[b, 0:32, n0:n0+256] -> LDS tile ----------
    // Each thread moves 8 x 16B; per-lane LDS byte offset in a VGPR, batch
    // base pointer uniform in an SGPR pair (GVS addressing).
    const float* fbase = feat + (size_t)b * VC * VN + n0;
#pragma unroll
    for (int r = 0; r < 8; ++r) {
        const int f   = r * 1024 + t * 4;        // flat float index in tile
        const int row = f >> 8;                  // channel
        const int col = f & 255;                 // point within block
        const unsigned voff = (unsigned)((row * VN + col) * 4);   // byte offset
        const unsigned lds  = (unsigned)(uintptr_t)&tile[f];      // LDS byte addr
        asm volatile("global_load_async_to_lds_b128 %0, %1, %2"
                     :: "v"(lds), "v"(voff), "s"(fbase)
                     : "memory");
    }

    // ---- voxel index for this thread's point (overlaps the async copy) -----
    const int n = n0 + t;
    const float* pb = pts + (size_t)b * 3 * VN;
    const float denom = sqrtf(__uint_as_float(*nmaxsq_bits)) + VEPS;
    const float x = pb[n]          - mins[b * 3 + 0];
    const float y = pb[VN + n]     - mins[b * 3 + 1];
    const float z = pb[2 * VN + n] - mins[b * 3 + 2];
    int ix = (int)((x / denom) * (float)(VR - 1));
    int iy = (int)((y / denom) * (float)(VR - 1));
    int iz = (int)((z / denom) * (float)(VR - 1));
    ix = min(max(ix, 0), VR - 1);
    iy = min(max(iy, 0), VR - 1);
    iz = min(max(iz, 0), VR - 1);
    const int vox = ix + (iy << 6) + (iz << 12);

    atomicAdd(&cnt[(size_t)b * VR3 + vox], 1u);

    // ---- wait for async tile, then scatter 32 channel atomics --------------
    asm volatile("s_wait_asynccnt 0" ::: "memory");
    __syncthreads();

    float* ob = out + (size_t)b * VC * VR3 + vox;
#pragma unroll 4
    for (int c = 0; c < VC; ++c) {
        // LDS read: lanes 0..31 hit distinct banks (stride 4B) -> conflict-free
        atomicAdd(ob + (size_t)c * VR3, tile[c * 256 + t]);
    }
}

// ---------------------------------------------------------------------------
// Kernel 4: out[b,c,v] /= max(cnt[b,v], 1).  262144 blocks x 256 threads.
// ---------------------------------------------------------------------------
__global__ __launch_bounds__(256) void k_final(float* __restrict__ out,
                                               const unsigned* __restrict__ cnt) {
    const size_t i = (size_t)blockIdx.x * 256 + threadIdx.x;  // < B*C*R^3
    const int v  = (int)(i & (VR3 - 1));
    const int bc = (int)(i >> 18);                            // b*C + c
    const int b  = bc >> 5;
    unsigned c = cnt[(size_t)b * VR3 + v];
    c = (c < 1u) ? 1u : c;
    out[i] = out[i] / (float)c;
}

// ---------------------------------------------------------------------------
extern "C" void kernel_launch(void* const* d_in, const int* in_sizes, int n_in,
                              void* d_out, int out_size, void* d_ws, size_t ws_size,
                              hipStream_t stream) {
    (void)in_sizes; (void)n_in; (void)ws_size;
    const float* pts  = (const float*)d_in[0];   // [8, 3, 65536]
    const float* feat = (const float*)d_in[1];   // [8, 32, 65536]
    float* out = (float*)d_out;                  // [8, 32, 64, 64, 64]

    // Workspace layout: [cnt: 8 MB][nmax: 4B ... pad][mins: 24 words]
    const size_t cnt_bytes = (size_t)VB * VR3 * 4;           // 8388608
    unsigned* cnt     = (unsigned*)d_ws;
    unsigned* nmax    = (unsigned*)((char*)d_ws + cnt_bytes);
    unsigned* minbits = (unsigned*)((char*)d_ws + cnt_bytes + 64);
    const float* minsf = (const float*)minbits;

    // Re-initialize accumulation state every call (graph-capturable memsets).
    hipMemsetAsync(d_out, 0, (size_t)out_size * sizeof(float), stream);
    hipMemsetAsync(d_ws, 0, cnt_bytes + 64, stream);          // cnt + nmax = 0
    hipMemsetAsync((void*)minbits, 0x7F, 24 * 4, stream);     // mins = 3.39e38f

    k_min    <<<VB * 3 * 32,       256, 0, stream>>>(pts, minbits);
    k_normmax<<<(VB * VN) / 256,   256, 0, stream>>>(pts, minsf, nmax);
    // Per-batch accumulation: 33 MB working set -> atomics stay in L2.
    for (int b = 0; b < VB; ++b) {
        k_accum<<<VN / 256, 256, 0, stream>>>(pts, feat, minsf, nmax, out, cnt, b);
    }
    k_final  <<<(VB * VC * VR3) / 256, 256, 0, stream>>>(out, cnt);
}